// Aperiodicity_25744033972311
// MI455X (gfx1250) — compile-verified
//
#include <hip/hip_runtime.h>
#include <math.h>

// ---------------------------------------------------------------------------
// Aperiodicity (WORLD-style band AP) for MI455X / gfx1250.
//
// Pipeline (all on `stream`, graph-capture safe):
//   1) qmf_kernel x3   : cascaded QMF analysis (reflect pad, decimate-by-2)
//   2) band_ap_kernel x4: one wave32 per (batch,frame); Gram matrix
//        M = G^T W G built with V_WMMA_F32_16X16X4_F32 accumulation over the
//        seg axis (K chunks of 4).  Window w[k] and sqrt(w[k]) are staged in
//        LDS once per block (they depend only on seg), zero-padded so the
//        K-tail needs no guards: A is identically 0 for k>=seg, annihilating
//        any garbage in B.  Inner loop = 2 ds loads + 2 global loads +
//        cndmask selects + 1 WMMA, fully branchless.
//   3) finalize_kernel : log-clip, piecewise-linear interp to 513 bins, exp.
//
// Problem is latency/bandwidth trivial (~1 GFLOP, ~50 MB traffic at 23.3TB/s)
// so the structure optimizes for coalesced streams + per-wave WMMA Gram.
// ---------------------------------------------------------------------------

typedef __attribute__((ext_vector_type(2))) float v2f;
typedef __attribute__((ext_vector_type(8))) float v8f;

#define NBATCH 8
#define T_IN   160000
#define NFRM   2000
#define NFTOT  (NBATCH * NFRM)   // 16000 frames total
#define NFREQ  513
#define SEGPAD 244               // max seg (241) rounded up to K-chunk multiple

// QMF prototype halves (full filters are palindromes of these)
__constant__ float AH[21] = {
    0.00041447996898231424f,  0.0007812505141729248f,  -0.0010917236836275842f,
   -0.001986792567596759f,    0.0020903896961562292f,   0.004094057027284935f,
   -0.0034025808529816698f,  -0.007496154127205602f,    0.004972263339933064f,
    0.012738791249119802f,   -0.006696032689574911f,   -0.020694051570247052f,
    0.008432436565041345f,    0.03307438375870053f,    -0.010018936738799522f,
   -0.05423136140580825f,     0.011293988915051487f,    0.10020081367388213f,
   -0.012120546202484579f,   -0.316300210390957f,       0.5124068258062764f };

__constant__ float AL[19] = {
   -0.0006548817007748305f,   7.561994958159384e-05f,   0.0020408456937895227f,
   -0.0007468053532203044f,  -0.004350223568826493f,    0.0025966428382642732f,
    0.007639602282756696f,   -0.006490411890149785f,   -0.011765804538954506f,
    0.013649908479276255f,    0.01636866479016021f,    -0.026075976030529347f,
   -0.020910294856659444f,    0.04826072503231665f,     0.02476784661104811f,
   -0.09617846758336064f,    -0.027359756709866623f,    0.3148805216163004f,
    0.5282734359405503f };

__device__ __forceinline__ float hhp_tap(int t) { return (t <= 20) ? AH[t] : AH[40 - t]; } // 41 taps
__device__ __forceinline__ float hlp_tap(int t) { return (t <= 18) ? AL[t] : AL[36 - t]; } // 37 taps

// ---------------------------------------------------------------------------
// QMF analysis step: y_hp[n] = sum_t xp[2n+t]*HHP[t] (pad 20, reflect)
//                    y_lp[n] = sum_t xp[2n+t]*HLP[t] (pad 18, reflect)
// ---------------------------------------------------------------------------
__global__ void qmf_kernel(const float* __restrict__ in, float* __restrict__ oh,
                           float* __restrict__ ol, int Tin)
{
  const int Tout = Tin >> 1;
  const long gid = (long)blockIdx.x * blockDim.x + threadIdx.x;
  if (gid >= (long)NBATCH * Tout) return;
  const int b = (int)(gid / Tout);
  const int n = (int)(gid - (long)b * Tout);
  const float* __restrict__ x = in + (long)b * Tin;

  const int base = 2 * n - 20;
  // hint the streaming access pattern into the cache hierarchy
  __builtin_prefetch(x + (base < 0 ? 0 : base) + 512, 0, 1);

  float xv[41];
#pragma unroll
  for (int t = 0; t < 41; ++t) {
    int i = base + t;
    i = (i < 0) ? -i : i;                   // reflect left
    i = (i >= Tin) ? (2 * Tin - 2 - i) : i; // reflect right
    xv[t] = x[i];
  }
  float yh = 0.f, yl = 0.f;
#pragma unroll
  for (int t = 0; t < 41; ++t) yh = fmaf(xv[t], hhp_tap(t), yh);
#pragma unroll
  for (int t = 0; t < 37; ++t) yl = fmaf(xv[t + 2], hlp_tap(t), yl);

  oh[(long)b * Tout + n] = yh;
  ol[(long)b * Tout + n] = yl;
}

// ---------------------------------------------------------------------------
// Per-band aperiodicity.  One wave32 per frame, 4 waves per block.
//
// G[k][col] (k = 0..seg-1):
//   col<3 : xb[clamp(origin - t0 + k + col - 1)]
//   col<6 : xb[clamp(origin + t0 + k + col - 4)]
//   col=6 : xb[clamp(origin + k)]
// w[k] = 0.5 - 0.5 cos(2*pi*(k+1)/(seg+1))    (hanning(seg+2)[1:-1])
//
// WMMA f32 16x16x4 accumulation:
//   A row m (m<7)  = w[k] * G[k][m]      -> D[m][n] = Gram M[m][n]
//   A row 7        = sqrt(w[k])          -> D[7][n] = sum sqrt(w)*G[:,n]
//   B col n (n<7)  = G[k][n]
// A layout (ISA 7.12.2, 16x4 f32): lane m=L&15, K = 2*(L>>4)+{0,1} in 2 VGPRs.
// B mirrored (row k striped across lanes).  Since m == n per lane, each lane
// does exactly one gather per K element.
// C/D layout: D[r][n] = VGPR r of the accumulator at lane n (lanes 0-15).
//
// Stats use (R+eps)a=b  =>  a^T R a = a.b - eps|a|^2:
//   resid_ss = Sxx - a.b - eps|a|^2 ,  sum_v = s_x - a.c
// ---------------------------------------------------------------------------
__global__ void __launch_bounds__(128)
band_ap_kernel(const float* __restrict__ xb, const float* __restrict__ f0,
               float* __restrict__ bap, int Tb, float tmp_fs, int seg)
{
  __shared__ float s_win[SEGPAD];
  __shared__ float s_sqw[SEGPAD];

  // Cooperative window fill (depends only on seg); zero-padded tail makes the
  // K-loop guard-free: A entries vanish for k >= seg.
  {
    const float wtheta = 6.283185307179586f / (float)(seg + 1);
    for (int t = threadIdx.x; t < SEGPAD; t += blockDim.x) {
      float w = (t < seg) ? (0.5f - 0.5f * cosf(wtheta * (float)(t + 1))) : 0.0f;
      s_win[t] = w;
      s_sqw[t] = sqrtf(w);
    }
  }
  __syncthreads();

  const int lane = threadIdx.x & 31;
  const int wid  = threadIdx.x >> 5;
  const int fidx = blockIdx.x * (blockDim.x >> 5) + wid;
  if (fidx >= NFTOT) return;                 // wave-uniform, after the barrier
  const int bt = fidx / NFRM;
  const int fr = fidx - bt * NFRM;

  float f0v = f0[fidx];
  if (f0v == 0.0f) f0v = 150.0f;             // UNVOICED -> DEFAULT_F0
  const float pitch = tmp_fs / f0v;
  const int t0     = (int)(pitch + 0.5f);
  const int bias   = (int)(pitch * 0.5f + 0.5f);
  const int curr   = (int)((float)fr * 0.005f * tmp_fs + 1.5f);
  const int origin = curr - bias;

  const float* __restrict__ xr = xb + (long)bt * Tb;
  const int T1 = Tb - 1;

  const int m  = lane & 15;                  // A row / B col owned by this lane
  const int kg = lane >> 4;                  // K half: lanes 0-15 K{0,1}, 16-31 K{2,3}

  // base gather offset for this lane's G column (cols 0..6; lanes 7-15 harmless)
  int gbase = origin;
  if (m < 3)       gbase = origin - t0 + m - 1;
  else if (m < 6)  gbase = origin + t0 + m - 4;

  const bool isG = (m < 7);                  // lane carries a real G column
  const bool isS = (m == 7);                 // lane carries the sqrt(w) row

  v8f acc = {};                              // 16x16 f32 accumulator (rows 0-7)
  const int nchunk = (seg + 3) >> 2;
  for (int c = 0; c < nchunk; ++c) {
    const int k0 = c * 4 + kg * 2;

    // window pair + sqrt-window pair from LDS (broadcast within lane halves)
    const float w0 = s_win[k0], w1 = s_win[k0 + 1];
    const float s0 = s_sqw[k0], s1 = s_sqw[k0 + 1];

    // gathers: always execute (clamped); garbage where w==0 is annihilated by A
    int i0 = gbase + k0;     i0 = i0 < 0 ? 0 : (i0 > T1 ? T1 : i0);
    int i1 = gbase + k0 + 1; i1 = i1 < 0 ? 0 : (i1 > T1 ? T1 : i1);
    const float g0 = xr[i0];
    const float g1 = xr[i1];

    // branchless operand assembly (cndmask chains, no EXEC divergence)
    v2f av, bv;
    float a0 = isG ? (w0 * g0) : 0.0f;  a0 = isS ? s0 : a0;
    float a1 = isG ? (w1 * g1) : 0.0f;  a1 = isS ? s1 : a1;
    av.x = a0; av.y = a1;
    bv.x = isG ? g0 : 0.0f;
    bv.y = isG ? g1 : 0.0f;

    acc = __builtin_amdgcn_wmma_f32_16x16x4_f32(false, av, false, bv,
                                                (short)0, acc, false, false);
  }

  // ---- pull M (rows 0-7, cols 0-6) out of the accumulator via shuffles ----
  float A6[6][7];    // [ R+eps | b ] augmented system
  float cvec[6], bvec[6], Sxx, sxw;
#pragma unroll
  for (int r = 0; r < 6; ++r) {
    float s = acc[r];
#pragma unroll
    for (int n = 0; n < 7; ++n) A6[r][n] = __shfl(s, n, 32);
  }
  { float s = acc[6]; Sxx = __shfl(s, 6, 32); }
  { float s = acc[7];
#pragma unroll
    for (int n = 0; n < 6; ++n) cvec[n] = __shfl(s, n, 32);
    sxw = __shfl(s, 6, 32);
  }
#pragma unroll
  for (int i = 0; i < 6; ++i) { bvec[i] = A6[i][6]; A6[i][i] += 1e-5f; }

  // ---- 6x6 solve, no pivoting (SPD + eps*I), redundant across lanes -------
  float a[6];
#pragma unroll
  for (int col = 0; col < 6; ++col) {
    const float inv = 1.0f / A6[col][col];
#pragma unroll
    for (int r = 0; r < 6; ++r) {
      if (r > col) {
        const float f = A6[r][col] * inv;
#pragma unroll
        for (int j = 0; j < 7; ++j)
          if (j >= col) A6[r][j] = fmaf(-f, A6[col][j], A6[r][j]);
      }
    }
  }
#pragma unroll
  for (int i = 5; i >= 0; --i) {
    float s = A6[i][6];
#pragma unroll
    for (int j = 0; j < 6; ++j)
      if (j > i) s = fmaf(-A6[i][j], a[j], s);
    a[i] = s / A6[i][i];
  }

  // ---- stds (ddof=1) from Gram-only quantities ----------------------------
  float adb = 0.f, asq = 0.f, adc = 0.f;
#pragma unroll
  for (int i = 0; i < 6; ++i) {
    adb = fmaf(a[i], bvec[i], adb);
    asq = fmaf(a[i], a[i], asq);
    adc = fmaf(a[i], cvec[i], adc);
  }
  const float resid_ss = Sxx - adb - 1e-5f * asq;   // sum (sqrt(w)*(X-Ha))^2
  const float sum_v    = sxw - adc;                 // sum  sqrt(w)*(X-Ha)
  const float invn1 = 1.0f / (float)(seg - 1);
  const float fseg  = (float)seg;
  const float var_n = (resid_ss - sum_v * sum_v / fseg) * invn1;
  const float var_d = (Sxx - sxw * sxw / fseg) * invn1;
  const float std_n = __builtin_sqrtf(fmaxf(var_n, 0.f));
  const float std_d = __builtin_sqrtf(fmaxf(var_d, 0.f));

  if (lane == 0) bap[fidx] = std_n / (std_d + 1e-16f);
}

// ---------------------------------------------------------------------------
// Final interpolation. bands reversed: [b3, b3, b2, b1, b0] over coarse freqs
// [0, 1000, 2000, 4000, 8000]; log-clip, linear interp, exp. 513 bins.
// ---------------------------------------------------------------------------
__device__ __forceinline__ float logclip(float v)
{
  v = fminf(fmaxf(v, 0.001f), 0.999f);
  return logf(v);
}

__global__ void finalize_kernel(const float* __restrict__ bap, float* __restrict__ out)
{
  const long gid = (long)blockIdx.x * blockDim.x + threadIdx.x;
  const long total = (long)NFTOT * NFREQ;
  if (gid >= total) return;
  const int fi = (int)(gid % NFREQ);
  const int bf = (int)(gid / NFREQ);

  const float yA = logclip(bap[3 * NFTOT + bf]);  // y[0] == y[1] (duplicated low band)
  const float yB = logclip(bap[2 * NFTOT + bf]);
  const float yC = logclip(bap[1 * NFTOT + bf]);
  const float yD = logclip(bap[0 * NFTOT + bf]);

  const float freq = (float)fi * 15.625f;         // SR / FFT_LENGTH
  int idx; float x0, dxv;
  if (freq <= 1000.f)      { idx = 0; x0 = 0.f;    dxv = 1000.f; }
  else if (freq <= 2000.f) { idx = 1; x0 = 1000.f; dxv = 1000.f; }
  else if (freq <= 4000.f) { idx = 2; x0 = 2000.f; dxv = 2000.f; }
  else                     { idx = 3; x0 = 4000.f; dxv = 4000.f; }
  const float w = (freq - x0) / dxv;

  const float y0 = (idx == 0) ? yA : (idx == 1) ? yA : (idx == 2) ? yB : yC;
  const float y1 = (idx == 0) ? yA : (idx == 1) ? yB : (idx == 2) ? yC : yD;

  out[gid] = expf((y1 - y0) * w + y0);
}

// ---------------------------------------------------------------------------
extern "C" void kernel_launch(void* const* d_in, const int* in_sizes, int n_in,
                              void* d_out, int out_size, void* d_ws, size_t ws_size,
                              hipStream_t stream)
{
  const float* x  = (const float*)d_in[0];   // (8, 160000) f32
  const float* f0 = (const float*)d_in[1];   // (8, 2000)   f32
  float* out = (float*)d_out;                // (8, 2000, 513) f32
  float* ws  = (float*)d_ws;

  // workspace layout (floats): ~9.2 MB total
  float* xb0 = ws;                               // 8 x 80000 (band 0, hp of x)
  float* lx1 = xb0 + (long)NBATCH * 80000;       // 8 x 80000
  float* xb1 = lx1 + (long)NBATCH * 80000;       // 8 x 40000 (band 1)
  float* lx2 = xb1 + (long)NBATCH * 40000;       // 8 x 40000
  float* xb2 = lx2 + (long)NBATCH * 40000;       // 8 x 20000 (band 2)
  float* lx3 = xb2 + (long)NBATCH * 20000;       // 8 x 20000 (band 3 = lowpass)
  float* bap = lx3 + (long)NBATCH * 20000;       // 4 x 16000

  // 1) QMF cascade
  {
    long n0 = (long)NBATCH * 80000;
    qmf_kernel<<<(int)((n0 + 255) / 256), 256, 0, stream>>>(x,   xb0, lx1, 160000);
    long n1 = (long)NBATCH * 40000;
    qmf_kernel<<<(int)((n1 + 255) / 256), 256, 0, stream>>>(lx1, xb1, lx2, 80000);
    long n2 = (long)NBATCH * 20000;
    qmf_kernel<<<(int)((n2 + 255) / 256), 256, 0, stream>>>(lx2, xb2, lx3, 40000);
  }

  // 2) per-band WMMA normal-equation solves: one wave per frame, 4 waves/block
  {
    const int WPB = 4;
    dim3 bsz(WPB * 32);
    const int nblk = (NFTOT + WPB - 1) / WPB;    // 4000 (exact, full waves)
    band_ap_kernel<<<nblk, bsz, 0, stream>>>(xb0, f0, bap + 0 * NFTOT, 80000, 8000.f, 241);
    band_ap_kernel<<<nblk, bsz, 0, stream>>>(xb1, f0, bap + 1 * NFTOT, 40000, 4000.f, 121);
    band_ap_kernel<<<nblk, bsz, 0, stream>>>(xb2, f0, bap + 2 * NFTOT, 20000, 2000.f,  61);
    band_ap_kernel<<<nblk, bsz, 0, stream>>>(lx3, f0, bap + 3 * NFTOT, 20000, 2000.f,  61);
  }

  // 3) interpolation to 513 bins
  {
    long nt = (long)NFTOT * NFREQ;
    finalize_kernel<<<(int)((nt + 255) / 256), 256, 0, stream>>>(bap, out);
  }
}